// OptNet_13039520711278
// MI455X (gfx1250) — compile-verified
//
#include <hip/hip_runtime.h>
#include <math.h>

// ---------------------------------------------------------------------------
// OptNet forward: batched QP interior-point solve + log_softmax, for gfx1250.
// One workgroup (256 threads = 8 wave32) per batch element; 20 IPM iterations
// fully in-kernel.  M = Q + G^T D G formed with V_WMMA_F32_16X16X4_F32,
// solved in LDS by unpivoted Gaussian elimination.
// ---------------------------------------------------------------------------

typedef __attribute__((ext_vector_type(2))) float v2f;
typedef __attribute__((ext_vector_type(8))) float v8f;

#define NN     128          // n = m = 128
#define ST     129          // LDS row stride (odd -> conflict-free columns)
#define ITERS  20
#define SIGMA  0.1f

// LDS layout (floats)
#define OFF_G    0
#define OFF_M    (NN * ST)                 // 16512
#define OFF_VEC  (2 * NN * ST)             // 33024
// 12 vectors of 128 floats each
#define OFF_Z    (OFF_VEC + 0 * NN)
#define OFF_S    (OFF_VEC + 1 * NN)
#define OFF_L    (OFF_VEC + 2 * NN)
#define OFF_RD   (OFF_VEC + 3 * NN)
#define OFF_RP   (OFF_VEC + 4 * NN)
#define OFF_D    (OFF_VEC + 5 * NN)
#define OFF_RC   (OFF_VEC + 6 * NN)
#define OFF_W    (OFF_VEC + 7 * NN)
#define OFF_RHS  (OFF_VEC + 8 * NN)
#define OFF_DZ   (OFF_VEC + 9 * NN)
#define OFF_DS   (OFF_VEC + 10 * NN)
#define OFF_DL   (OFF_VEC + 11 * NN)
#define OFF_RED  (OFF_VEC + 12 * NN)       // 256 floats
#define SMEM_FLOATS (OFF_RED + 256)        // 34816 floats = 139264 bytes

__global__ void optnet_ipm_kernel(const float* __restrict__ Qg,
                                  const float* __restrict__ pg,
                                  const float* __restrict__ Gg,
                                  const float* __restrict__ hg,
                                  float* __restrict__ zout) {
  extern __shared__ float smem[];
  float* Gs   = smem + OFF_G;
  float* Ms   = smem + OFF_M;
  float* zv   = smem + OFF_Z;
  float* sv   = smem + OFF_S;
  float* lamv = smem + OFF_L;
  float* rdv  = smem + OFF_RD;
  float* rpv  = smem + OFF_RP;
  float* dvec = smem + OFF_D;
  float* rcv  = smem + OFF_RC;
  float* wv   = smem + OFF_W;
  float* rhsv = smem + OFF_RHS;
  float* dzv  = smem + OFF_DZ;
  float* dsv  = smem + OFF_DS;
  float* dlv  = smem + OFF_DL;
  float* red  = smem + OFF_RED;

  const int tid  = threadIdx.x;
  const int b    = blockIdx.x;
  const int lane = tid & 31;
  const int wave = tid >> 5;           // 0..7 -> tile row of M
  const int m16  = lane & 15;
  const int khalf = (lane >> 4) << 1;  // 0 for lanes 0-15, 2 for lanes 16-31
  const int rowoff = (lane >> 4) * 8;  // C/D layout: upper lanes hold M+8

  // --- load shared G into LDS; init state z=0, s=1, lam=1 -------------------
  for (int idx = tid; idx < NN * NN; idx += 256)
    Gs[(idx >> 7) * ST + (idx & 127)] = Gg[idx];
  if (tid < NN) { zv[tid] = 0.0f; sv[tid] = 1.0f; lamv[tid] = 1.0f; }
  __syncthreads();

  for (int it = 0; it < ITERS; ++it) {
    // --- residuals: r_d = Qz + p + G^T lam ; r_p = Gz + s - h ---------------
    if (tid < NN) {
      float a1 = pg[b * NN + tid];
      for (int j = 0; j < NN; ++j) a1 += Qg[tid * NN + j] * zv[j];
      for (int k = 0; k < NN; ++k) a1 += lamv[k] * Gs[k * ST + tid];
      rdv[tid] = a1;
      float a2 = sv[tid] - hg[tid];
      for (int j = 0; j < NN; ++j) a2 += Gs[tid * ST + j] * zv[j];
      rpv[tid] = a2;
    }
    // mu = mean(s*lam)
    red[tid] = (tid < NN) ? sv[tid] * lamv[tid] : 0.0f;
    __syncthreads();
    for (int sh = 128; sh > 0; sh >>= 1) {
      if (tid < sh) red[tid] += red[tid + sh];
      __syncthreads();
    }
    const float mu = red[0] * (1.0f / 128.0f);
    __syncthreads();

    // --- centering terms ----------------------------------------------------
    if (tid < NN) {
      float si = sv[tid], li = lamv[tid];
      dvec[tid] = li / si;
      float rc = si * li - SIGMA * mu;
      rcv[tid] = rc;
      wv[tid] = (rc - li * rpv[tid]) / si;
    }
    __syncthreads();

    // rhs = -r_d + G^T w
    if (tid < NN) {
      float a3 = -rdv[tid];
      for (int k = 0; k < NN; ++k) a3 += wv[k] * Gs[k * ST + tid];
      rhsv[tid] = a3;
    }
    __syncthreads();

    // --- M = Q + G^T diag(d) G via V_WMMA_F32_16X16X4_F32 -------------------
    {
      const int ti = wave;               // 16-row strip of M
      v8f acc[8];
#pragma unroll
      for (int tj = 0; tj < 8; ++tj) {
#pragma unroll
        for (int r = 0; r < 8; ++r)
          acc[tj][r] = Qg[(ti * 16 + r + rowoff) * NN + tj * 16 + m16];
      }
#pragma unroll 4
      for (int kk = 0; kk < 32; ++kk) {
        const int k0 = kk * 4 + khalf;
        v2f a;                           // A[m,k] = G[k, ti*16+m] * d[k]
        a.x = Gs[k0 * ST + ti * 16 + m16] * dvec[k0];
        a.y = Gs[(k0 + 1) * ST + ti * 16 + m16] * dvec[k0 + 1];
#pragma unroll
        for (int tj = 0; tj < 8; ++tj) {
          v2f bb;                        // B[k,j] = G[k, tj*16+j]
          bb.x = Gs[k0 * ST + tj * 16 + m16];
          bb.y = Gs[(k0 + 1) * ST + tj * 16 + m16];
          acc[tj] = __builtin_amdgcn_wmma_f32_16x16x4_f32(
              false, a, false, bb, (short)0, acc[tj], false, false);
        }
      }
#pragma unroll
      for (int tj = 0; tj < 8; ++tj) {
#pragma unroll
        for (int r = 0; r < 8; ++r)
          Ms[(ti * 16 + r + rowoff) * ST + tj * 16 + m16] = acc[tj][r];
      }
    }
    __syncthreads();

    // augmented column = rhs
    if (tid < NN) Ms[tid * ST + NN] = rhsv[tid];
    __syncthreads();

    // --- Gaussian elimination (SPD, no pivoting) on 128x129 in LDS ----------
    for (int k = 0; k < NN - 1; ++k) {
      const float inv = 1.0f / Ms[k * ST + k];
      const int r = k + 1 + (tid >> 1);
      if (r < NN) {
        const float factor = Ms[r * ST + k] * inv;
        for (int c = k + 1 + (tid & 1); c <= NN; c += 2)
          Ms[r * ST + c] -= factor * Ms[k * ST + c];
      }
      __syncthreads();
    }
    // back substitution (column sweep)
    for (int k = NN - 1; k >= 0; --k) {
      if (tid == 0) dzv[k] = Ms[k * ST + NN] / Ms[k * ST + k];
      __syncthreads();
      if (tid < k) Ms[tid * ST + NN] -= Ms[tid * ST + k] * dzv[k];
      __syncthreads();
    }

    // --- ds = -r_p - G dz ; dlam = (-r_c - lam*ds)/s ------------------------
    if (tid < NN) {
      float a4 = -rpv[tid];
      for (int j = 0; j < NN; ++j) a4 -= Gs[tid * ST + j] * dzv[j];
      dsv[tid] = a4;
      dlv[tid] = (-rcv[tid] - lamv[tid] * a4) / sv[tid];
    }
    __syncthreads();

    // --- fraction-to-boundary step length -----------------------------------
    float rat = 1e9f;
    if (tid < NN) {
      const float dsi = dsv[tid], dli = dlv[tid];
      if (dsi < 0.0f) rat = fminf(rat, -sv[tid] / dsi);
      if (dli < 0.0f) rat = fminf(rat, -lamv[tid] / dli);
    }
    red[tid] = rat;
    __syncthreads();
    for (int sh = 128; sh > 0; sh >>= 1) {
      if (tid < sh) red[tid] = fminf(red[tid], red[tid + sh]);
      __syncthreads();
    }
    const float alpha = fminf(1.0f, 0.99f * red[0]);
    __syncthreads();

    if (tid < NN) {
      zv[tid]   += alpha * dzv[tid];
      sv[tid]   += alpha * dsv[tid];
      lamv[tid] += alpha * dlv[tid];
    }
    __syncthreads();
  }

  if (tid < NN) zout[b * NN + tid] = zv[tid];
}

// ---------------------------------------------------------------------------
// log_softmax over rows of the (10, 32768) reshaped output, in place.
// ---------------------------------------------------------------------------
__global__ void optnet_lsm_kernel(float* __restrict__ out) {
  const int ROW = 32768;
  float* v = out + blockIdx.x * ROW;
  const int tid = threadIdx.x;
  __shared__ float red[256];

  float mx = -INFINITY;
  for (int i = tid; i < ROW; i += 256) mx = fmaxf(mx, v[i]);
  red[tid] = mx;
  __syncthreads();
  for (int sh = 128; sh > 0; sh >>= 1) {
    if (tid < sh) red[tid] = fmaxf(red[tid], red[tid + sh]);
    __syncthreads();
  }
  mx = red[0];
  __syncthreads();

  float sum = 0.0f;
  for (int i = tid; i < ROW; i += 256) sum += expf(v[i] - mx);
  red[tid] = sum;
  __syncthreads();
  for (int sh = 128; sh > 0; sh >>= 1) {
    if (tid < sh) red[tid] += red[tid + sh];
    __syncthreads();
  }
  const float lse = mx + logf(red[0]);

  for (int i = tid; i < ROW; i += 256) v[i] -= lse;
}

extern "C" void kernel_launch(void* const* d_in, const int* in_sizes, int n_in,
                              void* d_out, int out_size, void* d_ws, size_t ws_size,
                              hipStream_t stream) {
  (void)in_sizes; (void)n_in; (void)d_ws; (void)ws_size; (void)out_size;
  // setup_inputs order: x, Q, p, G, h, m   (x and m unused by the reference path)
  const float* Q = (const float*)d_in[1];
  const float* p = (const float*)d_in[2];
  const float* G = (const float*)d_in[3];
  const float* h = (const float*)d_in[4];
  float* out = (float*)d_out;

  constexpr size_t smem_bytes = (size_t)SMEM_FLOATS * sizeof(float); // 139264
  (void)hipFuncSetAttribute((const void*)optnet_ipm_kernel,
                            hipFuncAttributeMaxDynamicSharedMemorySize,
                            (int)smem_bytes);

  optnet_ipm_kernel<<<2560, 256, smem_bytes, stream>>>(Q, p, G, h, out);
  optnet_lsm_kernel<<<10, 256, 0, stream>>>(out);
}